// PsdFeatureExtractor_32607391711296
// MI455X (gfx1250) — compile-verified
//
#include <hip/hip_runtime.h>

typedef float v2f __attribute__((ext_vector_type(2)));
typedef float v8f __attribute__((ext_vector_type(8)));

#define SIG_LEN     4000
#define PAD         100
#define XP_LEN      4200      // reflect-padded length
#define N_FRAMES    501       // 4000/8 + 1
#define HOP         8
#define N_BINS      100       // bands cover bins 0..99 only
#define BIN_TILES   7         // 7*16 = 112 >= 100
#define FRAME_TILES 32        // 32*16 = 512 >= 501
#define N_BANDS     7
#define T_OUT       501
#define NWAVES      8

__device__ __forceinline__ int band_of(int b) {
    // (0,4)(4,8)(8,12)(12,30)(30,50)(50,70)(70,100)
    if (b < 12) return b >> 2;
    if (b < 30) return 3;
    if (b < 50) return 4;
    if (b < 70) return 5;
    return 6;
}

__device__ __forceinline__ float inv_band_size(int bd) {
    if (bd < 3)  return 0.25f;
    if (bd == 3) return (1.0f / 18.0f);
    if (bd == 6) return (1.0f / 30.0f);
    return 0.05f;              // bands 4,5: width 20
}

__global__ void psd_zero_kernel(float* __restrict__ out, int n) {
    int i = blockIdx.x * blockDim.x + threadIdx.x;
    if (i < n) out[i] = 0.0f;
}

__global__ __launch_bounds__(256) void psd_feature_kernel(const float* __restrict__ x,
                                                          float* __restrict__ out) {
    __shared__ __align__(16) float s_x[XP_LEN];

    const int sig   = blockIdx.x;
    const int tid   = threadIdx.x;
    const int lane  = tid & 31;
    const int wave  = tid >> 5;
    const int hi    = lane >> 4;     // half-wave select (K/M high half)
    const int lan16 = lane & 15;

    // Stage reflect-padded signal into LDS (coalesced except 200 edge taps).
    const float* xs = x + (size_t)sig * SIG_LEN;
    for (int i = tid; i < XP_LEN; i += 256) {
        int j = i - PAD;
        j = (j < 0) ? -j : j;
        j = (j >= SIG_LEN) ? (2 * SIG_LEN - 2 - j) : j;
        s_x[i] = xs[j];
    }
    __syncthreads();

    float* out_sig = out + (size_t)sig * N_BANDS * T_OUT;

    v2f are[4], aim[4];
    int cur_bt = -1;

    const int n_tasks = BIN_TILES * FRAME_TILES;   // 224
    for (int task = wave; task < n_tasks; task += NWAVES) {
        const int bt = task >> 5;    // bin tile 0..6  (uniform per wave)
        const int ft = task & 31;    // frame tile 0..31
        const int m0 = bt << 4;

        // Rebuild A operands (windowed DFT coefficients) when the bin tile changes.
        // A layout (16x4 f32): lane L holds K = v + 2*(L>=16), M = L%16.
        if (bt != cur_bt) {
            cur_bt = bt;
            const int bin = m0 + lan16;
#pragma unroll
            for (int s = 0; s < 4; ++s) {
#pragma unroll
                for (int v = 0; v < 2; ++v) {
                    const int tap = 4 * s + v + 2 * hi;        // K index 0..15
                    const float w =
                        0.5f * (1.0f - __cosf((float)tap * (6.28318530717958647f / 16.0f)));
                    const int ph = (bin * (92 + tap)) % 200;   // exact phase reduction
                    const float th = (float)ph * (6.28318530717958647f / 200.0f);
                    float sn, cs;
                    __sincosf(th, &sn, &cs);
                    are[s][v] = w * cs;   // Re coefficient
                    aim[s][v] = w * sn;   // Im coefficient (sign irrelevant: squared)
                }
            }
        }

        // B operand from LDS: lane L supplies K = v + 2*hi, N(frame) = ft*16 + L%16.
        // Taps v=0,1 are adjacent in LDS -> one aligned 8-byte ds load per K-step.
        const int frame = (ft << 4) + lan16;
        const int fcl   = frame < N_FRAMES ? frame : (N_FRAMES - 1);
        const int base  = fcl * HOP + 92 + 2 * hi;

        v8f cre = {0.f, 0.f, 0.f, 0.f, 0.f, 0.f, 0.f, 0.f};
        v8f cim = {0.f, 0.f, 0.f, 0.f, 0.f, 0.f, 0.f, 0.f};
#pragma unroll
        for (int s = 0; s < 4; ++s) {
            const v2f b = *(const v2f*)(s_x + base + 4 * s);
            cre = __builtin_amdgcn_wmma_f32_16x16x4_f32(false, are[s], false, b,
                                                        (short)0, cre, false, false);
            cim = __builtin_amdgcn_wmma_f32_16x16x4_f32(false, aim[s], false, b,
                                                        (short)0, cim, false, false);
        }

        // D layout: element v of each lane is bin m0 + v + 8*hi, frame = ft*16 + L%16.
        // Bands are monotone in bin -> run-length accumulate, flush per band change.
        if (frame < N_FRAMES) {
            float* op = out_sig + frame;
            int curb = -1;
            float acc = 0.0f;
#pragma unroll
            for (int v = 0; v < 8; ++v) {
                const int b = m0 + v + 8 * hi;
                if (b < N_BINS) {
                    const float re  = cre[v];
                    const float im  = cim[v];
                    const float amp = __logf(re * re + im * im + 1e-10f);
                    const int bd = band_of(b);
                    if (bd != curb) {
                        if (curb >= 0)
                            unsafeAtomicAdd(op + curb * T_OUT, acc * inv_band_size(curb));
                        curb = bd;
                        acc  = 0.0f;
                    }
                    acc += amp;
                }
            }
            if (curb >= 0)
                unsafeAtomicAdd(op + curb * T_OUT, acc * inv_band_size(curb));
        }
    }
}

extern "C" void kernel_launch(void* const* d_in, const int* in_sizes, int n_in,
                              void* d_out, int out_size, void* d_ws, size_t ws_size,
                              hipStream_t stream) {
    const float* x = (const float*)d_in[0];
    float* out = (float*)d_out;

    const int n_signals = in_sizes[0] / SIG_LEN;   // 32*32 = 1024

    // Output is accumulated with atomics -> zero it first (harness poisons buffers).
    psd_zero_kernel<<<(out_size + 255) / 256, 256, 0, stream>>>(out, out_size);

    // One block per signal: 256 threads = 8 wave32s, signal staged in 16.8 KB LDS.
    psd_feature_kernel<<<n_signals, 256, 0, stream>>>(x, out);
}